// SSA_86861418594902
// MI455X (gfx1250) — compile-verified
//
#include <hip/hip_runtime.h>

typedef __attribute__((ext_vector_type(2))) float v2f;
typedef __attribute__((ext_vector_type(8))) float v8f;
typedef unsigned int u32x4 __attribute__((ext_vector_type(4)));
typedef int          i32x8 __attribute__((ext_vector_type(8)));
typedef int          i32x4 __attribute__((ext_vector_type(4)));

#define B_      2
#define N_      8192
#define D_      64
#define NEGV    (-1.0e10f)
#define SCALE_  0.125f          // 1/sqrt(64)
#define MWORDS  (N_ / 32)       // 256 mask words per row
#define WAVES_  4
#define CTILES  (N_ / 16)       // 512 column tiles
#define TPW     (CTILES / WAVES_)

__device__ __forceinline__ v8f wmma_f32(v2f a, v2f b, v8f c) {
  // D = A(16x4 f32) * B(4x16 f32) + C(16x16 f32)
  return __builtin_amdgcn_wmma_f32_16x16x4_f32(
      /*neg_a=*/false, a, /*neg_b=*/false, b,
      /*c_mod=*/(short)0, c, /*reuse_a=*/false, /*reuse_b=*/false);
}

// TDM: 1-D contiguous DMA of nElem f32 from global into LDS (per-wave issue,
// EXEC ignored, tracked by TENSORcnt). D# per CDNA5 ISA §8 (CDNA5 ISA).
// This toolchain exposes the 6-arg builtin (extra i32x8 group before cpol).
__device__ __forceinline__ void tdm_load_lin(float* lds_dst, const float* gsrc,
                                             int nElem) {
  unsigned lds_addr = (unsigned)(uintptr_t)lds_dst;  // low 32b of flat = LDS offset
  unsigned long long ga = (unsigned long long)(uintptr_t)gsrc;
  u32x4 g0 = { 1u,                                   // count=1, user mode
               lds_addr,                             // lds_addr [63:32]
               (unsigned)ga,                         // global_addr lo
               ((unsigned)(ga >> 32) & 0x01FFFFFFu)  // global_addr[56:32]
                   | (2u << 30) };                   // type=2 ("image")
  i32x8 g1 = { (int)(2u << 16),                      // data_size=2 -> 4 bytes
               (int)((unsigned)(nElem & 0xFFFF) << 16),          // tensor_dim0 lo16
               (int)(((unsigned)nElem >> 16) | (1u << 16)),      // dim0 hi16 | tensor_dim1=1
               (int)((unsigned)(nElem & 0xFFFF) << 16),          // dim1 hi=0 | tile_dim0
               1,                                    // tile_dim1=1, tile_dim2=0
               nElem,                                // tensor_dim0_stride lo32
               0, 0 };                               // stride hi / dim1_stride
  i32x4 gz4 = {0, 0, 0, 0};
  i32x8 gz8 = {0, 0, 0, 0, 0, 0, 0, 0};
  __builtin_amdgcn_tensor_load_to_lds(g0, g1, gz4, gz4, gz8, 0);
}

// ---------------- QKV projection: y = x @ W^T (+ b for V) ----------------
__global__ void SSA_proj(const float* __restrict__ x,
                         const float* __restrict__ Wq,
                         const float* __restrict__ Wk,
                         const float* __restrict__ Wv,
                         const float* __restrict__ bv,
                         float* __restrict__ q, float* __restrict__ k,
                         float* __restrict__ v) {
  int gid = blockIdx.x * blockDim.x + threadIdx.x;   // B*N*D threads
  int row = gid >> 6;
  int d   = gid & 63;
  const float* xr = x + (size_t)row * D_;
  const float* wq = Wq + d * D_;
  const float* wk = Wk + d * D_;
  const float* wv = Wv + d * D_;
  float sq = 0.f, sk = 0.f, sv = 0.f;
#pragma unroll 8
  for (int i = 0; i < D_; ++i) {
    float xv = xr[i];
    sq = fmaf(xv, wq[i], sq);
    sk = fmaf(xv, wk[i], sk);
    sv = fmaf(xv, wv[i], sv);
  }
  q[gid] = sq;
  k[gid] = sk;
  v[gid] = sv + bv[d];
}

// ---------------- mask bitmap build ----------------
__global__ void SSA_mask_zero(unsigned* __restrict__ mw) {
  int i = blockIdx.x * blockDim.x + threadIdx.x;
  if (i < N_ * MWORDS) mw[i] = 0u;
}

__global__ void SSA_mask_scatter(const int* __restrict__ edge, int E,
                                 unsigned* __restrict__ mw) {
  int i = blockIdx.x * blockDim.x + threadIdx.x;
  if (i >= E) return;
  int r = edge[i];
  int c = edge[E + i];
  atomicOr(&mw[(size_t)r * MWORDS + (c >> 5)], 1u << (c & 31));
}

// ---------------- fused masked attention ----------------
// grid = B * N/16 blocks of 128 threads (4 waves); 16 query rows per block.
// Each wave owns TPW column tiles. K/V tiles arrive via double-buffered TDM
// DMA (tensor_load_to_lds); exact two-pass online softmax; attn written once.
__global__ void __launch_bounds__(128)
SSA_attn(const float* __restrict__ qg, const float* __restrict__ kg,
         const float* __restrict__ vg, const unsigned* __restrict__ maskw,
         float* __restrict__ outp, float* __restrict__ attnp) {
  __shared__ float    q_lds[16 * 64];                 // 4KB  (block-shared)
  __shared__ unsigned mask_lds[16 * MWORDS];          // 16KB (block-shared)
  __shared__ float    k_lds[WAVES_ * 2 * 16 * 64];    // 32KB (per-wave dbl buf)
  __shared__ float    v_lds[WAVES_ * 2 * 16 * 64];    // 32KB
  __shared__ float    p_lds[WAVES_ * 16 * 16];        // 4KB
  __shared__ float    st_m[WAVES_ * 16];
  __shared__ float    st_l[WAVES_ * 16];

  const int tid  = threadIdx.x;
  const int wave = tid >> 5;
  const int lane = tid & 31;
  const int r    = lane & 15;   // col-in-tile (B/C/D) or row (A operand)
  const int h    = lane >> 4;   // lane half -> K/M offset
  const int bIdx = blockIdx.x / (N_ / 16);
  const int row0 = (blockIdx.x % (N_ / 16)) * 16;

  // ---- cooperative loads: Q rows + mask rows ----
  {
    const float4* src = (const float4*)(qg + ((size_t)bIdx * N_ + row0) * D_);
    float4* dst = (float4*)q_lds;
    for (int i = tid; i < 256; i += 128) dst[i] = src[i];
    const uint4* ms = (const uint4*)(maskw + (size_t)row0 * MWORDS);
    uint4* md = (uint4*)mask_lds;
    for (int i = tid; i < (16 * MWORDS) / 4; i += 128) md[i] = ms[i];
  }
  __syncthreads();

  float* kb0 = k_lds + wave * 2048;
  float* kb1 = kb0 + 1024;
  float* vb0 = v_lds + wave * 2048;
  float* vb1 = vb0 + 1024;
  float* myp = p_lds + wave * 256;
  const int t0 = wave * TPW, t1 = t0 + TPW;
  const float* kbase = kg + (size_t)bIdx * N_ * D_;
  const float* vbase = vg + (size_t)bIdx * N_ * D_;

  float m_l[8], l_l[8];
#pragma unroll
  for (int g = 0; g < 8; ++g) { m_l[g] = -3.0e38f; l_l[g] = 0.f; }

  // =================== pass 1: row max + denom ===================
  tdm_load_lin(kb0, kbase + (size_t)t0 * 16 * D_, 1024);  // prologue DMA
  for (int t = t0; t < t1; ++t) {
    float* cur = ((t - t0) & 1) ? kb1 : kb0;
    asm volatile("s_wait_dscnt 0" ::: "memory");  // reads of other buf done
    if (t + 1 < t1) {
      float* nxt = ((t + 1 - t0) & 1) ? kb1 : kb0;
      tdm_load_lin(nxt, kbase + (size_t)(t + 1) * 16 * D_, 1024);
      __builtin_amdgcn_s_wait_tensorcnt(1);       // tile t landed, t+1 in flight
    } else {
      __builtin_amdgcn_s_wait_tensorcnt(0);
    }

    v8f acc = {0.f, 0.f, 0.f, 0.f, 0.f, 0.f, 0.f, 0.f};
#pragma unroll
    for (int kk = 0; kk < 64; kk += 4) {
      v2f a = *(const v2f*)(q_lds + r * 64 + kk + 2 * h);
      v2f b = *(const v2f*)(cur + r * 64 + kk + 2 * h);
      acc = wmma_f32(a, b, acc);
    }
    const int c0   = t * 16;
    const int wsel = c0 >> 5;
    const int bitb = (c0 & 31) + r;
#pragma unroll
    for (int g = 0; g < 8; ++g) {
      const int mrow = g + 8 * h;
      unsigned mw = mask_lds[mrow * MWORDS + wsel];
      float s  = ((mw >> bitb) & 1u) ? acc[g] * SCALE_ : NEGV;
      float mn = fmaxf(m_l[g], s);
      l_l[g] = l_l[g] * __expf(m_l[g] - mn) + __expf(s - mn);
      m_l[g] = mn;
    }
  }

  // merge 16 lanes of each half (butterfly), publish per wave, merge waves
#pragma unroll
  for (int g = 0; g < 8; ++g) {
#pragma unroll
    for (int off = 1; off <= 8; off <<= 1) {
      float mo = __shfl_xor(m_l[g], off, 32);
      float lo = __shfl_xor(l_l[g], off, 32);
      float mn = fmaxf(m_l[g], mo);
      l_l[g] = l_l[g] * __expf(m_l[g] - mn) + lo * __expf(mo - mn);
      m_l[g] = mn;
    }
  }
#pragma unroll
  for (int g = 0; g < 8; ++g)
    if (r == g) { st_m[wave * 16 + g + 8 * h] = m_l[g];
                  st_l[wave * 16 + g + 8 * h] = l_l[g]; }
  __syncthreads();
#pragma unroll
  for (int g = 0; g < 8; ++g) {
    const int mrow = g + 8 * h;
    float mf = -3.0e38f;
#pragma unroll
    for (int w2 = 0; w2 < WAVES_; ++w2) mf = fmaxf(mf, st_m[w2 * 16 + mrow]);
    float lf = 0.f;
#pragma unroll
    for (int w2 = 0; w2 < WAVES_; ++w2)
      lf += st_l[w2 * 16 + mrow] * __expf(st_m[w2 * 16 + mrow] - mf);
    m_l[g] = mf;
    l_l[g] = 1.0f / lf;   // store reciprocal denom
  }

  // =================== pass 2: write attn + O += P*V ===================
  v8f accO[4];
#pragma unroll
  for (int nb = 0; nb < 4; ++nb)
    accO[nb] = (v8f){0.f, 0.f, 0.f, 0.f, 0.f, 0.f, 0.f, 0.f};

  tdm_load_lin(kb0, kbase + (size_t)t0 * 16 * D_, 1024);
  tdm_load_lin(vb0, vbase + (size_t)t0 * 16 * D_, 1024);
  for (int t = t0; t < t1; ++t) {
    const int buf = (t - t0) & 1;
    float* curk = buf ? kb1 : kb0;
    float* curv = buf ? vb1 : vb0;
    asm volatile("s_wait_dscnt 0" ::: "memory");
    if (t + 1 < t1) {
      const int nb2 = (t + 1 - t0) & 1;
      tdm_load_lin(nb2 ? kb1 : kb0, kbase + (size_t)(t + 1) * 16 * D_, 1024);
      tdm_load_lin(nb2 ? vb1 : vb0, vbase + (size_t)(t + 1) * 16 * D_, 1024);
      __builtin_amdgcn_s_wait_tensorcnt(2);       // this tile's K+V landed
    } else {
      __builtin_amdgcn_s_wait_tensorcnt(0);
    }

    v8f acc = {0.f, 0.f, 0.f, 0.f, 0.f, 0.f, 0.f, 0.f};
#pragma unroll
    for (int kk = 0; kk < 64; kk += 4) {
      v2f a = *(const v2f*)(q_lds + r * 64 + kk + 2 * h);
      v2f b = *(const v2f*)(curk + r * 64 + kk + 2 * h);
      acc = wmma_f32(a, b, acc);
    }
    const int c0   = t * 16;
    const int wsel = c0 >> 5;
    const int bitb = (c0 & 31) + r;
#pragma unroll
    for (int g = 0; g < 8; ++g) {
      const int mrow = g + 8 * h;
      unsigned mw = mask_lds[mrow * MWORDS + wsel];
      float s = ((mw >> bitb) & 1u) ? acc[g] * SCALE_ : NEGV;
      float p = __expf(s - m_l[g]) * l_l[g];          // normalized prob
      attnp[((size_t)bIdx * N_ + row0 + mrow) * N_ + c0 + r] = p;
      myp[mrow * 16 + r] = p;
    }
    asm volatile("s_wait_dscnt 0" ::: "memory");

    // O[16x64] += P[16x16] @ V[16x64], 4 column blocks of 16
#pragma unroll
    for (int nb = 0; nb < 4; ++nb) {
      const int d0 = nb * 16;
#pragma unroll
      for (int kk = 0; kk < 16; kk += 4) {
        v2f a = *(const v2f*)(myp + r * 16 + kk + 2 * h);
        v2f b;
        b.x = curv[(kk + 2 * h) * 64 + d0 + r];
        b.y = curv[(kk + 2 * h + 1) * 64 + d0 + r];
        accO[nb] = wmma_f32(a, b, accO[nb]);
      }
    }
  }

  // merge partial O across the 4 waves (reuse k_lds as scratch)
  __syncthreads();   // all waves done with their K/V double buffers
  {
    float* osc = k_lds + wave * 1024;
#pragma unroll
    for (int nb = 0; nb < 4; ++nb)
#pragma unroll
      for (int g = 0; g < 8; ++g)
        osc[(g + 8 * h) * 64 + nb * 16 + r] = accO[nb][g];
  }
  __syncthreads();
  {
    float* op = outp + ((size_t)bIdx * N_ + row0) * D_;
    for (int i = tid; i < 1024; i += 128)
      op[i] = k_lds[i] + k_lds[1024 + i] + k_lds[2048 + i] + k_lds[3072 + i];
  }
}

extern "C" void kernel_launch(void* const* d_in, const int* in_sizes, int n_in,
                              void* d_out, int out_size, void* d_ws, size_t ws_size,
                              hipStream_t stream) {
  const float* x  = (const float*)d_in[0];
  const float* Wq = (const float*)d_in[1];
  const float* Wk = (const float*)d_in[2];
  const float* Wv = (const float*)d_in[3];
  const float* bv = (const float*)d_in[4];
  const int*   ed = (const int*)d_in[5];
  const int E = in_sizes[5] / 2;

  float* q = (float*)d_ws;                       // 4MB
  float* k = q + (size_t)B_ * N_ * D_;           // 4MB
  float* v = k + (size_t)B_ * N_ * D_;           // 4MB
  unsigned* mw = (unsigned*)(v + (size_t)B_ * N_ * D_);  // 8MB bitmap

  float* outp  = (float*)d_out;
  float* attnp = outp + (size_t)B_ * N_ * D_;

  SSA_proj<<<(B_ * N_ * D_) / 256, 256, 0, stream>>>(x, Wq, Wk, Wv, bv, q, k, v);
  SSA_mask_zero<<<(N_ * MWORDS + 255) / 256, 256, 0, stream>>>(mw);
  SSA_mask_scatter<<<(E + 255) / 256, 256, 0, stream>>>(ed, E, mw);
  SSA_attn<<<B_ * (N_ / 16), 128, 0, stream>>>(q, k, v, mw, outp, attnp);
}